// meandeterminant_3728031613735
// MI455X (gfx1250) — compile-verified
//
#include <hip/hip_runtime.h>

typedef __attribute__((ext_vector_type(4))) float f4;
typedef __attribute__((ext_vector_type(4))) int v4i;

#define BLOCK 256
#define ROWS_PER_THREAD 4
#define ROWS_PER_BLOCK (BLOCK * ROWS_PER_THREAD)   // 1024 rows
#define FLOATS_PER_BLOCK (ROWS_PER_BLOCK * 9)      // 9216 floats = 36 KB
#define ROWS_PER_WAVE (32 * ROWS_PER_THREAD)       // 128 rows
#define SLOTS_PER_WAVE (ROWS_PER_WAVE * 9 / 4)     // 288 float4 slots
#define VEC4_PER_LANE (SLOTS_PER_WAVE / 32)        // 9 copies per lane

__device__ __forceinline__ float det3x3(const float* m) {
    float a = m[0], b = m[1], c = m[2];
    float d = m[3], e = m[4], f = m[5];
    float g = m[6], h = m[7], i = m[8];
    return a * (e * i - f * h) - b * (d * i - f * g) + c * (d * h - e * g);
}

__global__ __launch_bounds__(BLOCK) void det3_async_kernel(
    const float* __restrict__ x, float* __restrict__ out, long long nrows) {
    __shared__ __align__(16) float lds[FLOATS_PER_BLOCK];

    const int t = threadIdx.x;
    const long long blockRow0 = (long long)blockIdx.x * ROWS_PER_BLOCK;

    if (blockRow0 + ROWS_PER_BLOCK <= nrows) {
        // ---------- fast path: full 1024-row tile, per-wave sub-tiles ----------
        const int lane = t & 31;
        const int wave = t >> 5;

        // Wave w owns a contiguous 288-float4 (1152-float) region; its lanes
        // both load and consume only that region, so a wave-local
        // s_wait_asynccnt is the only synchronization needed (no __syncthreads).
        const float* gwave =
            x + blockRow0 * 9 + (long long)wave * (SLOTS_PER_WAVE * 4);
        float* lwave = &lds[wave * (SLOTS_PER_WAVE * 4)];

        // CDNA5 async global->LDS b128 copies: each instruction moves a
        // contiguous 512B per wave straight into LDS (no VGPR staging),
        // tracked by ASYNCcnt.
#pragma unroll
        for (int i = 0; i < VEC4_PER_LANE; ++i) {
            const int s = lane + i * 32;  // float4 slot within wave region
            __builtin_amdgcn_global_load_async_to_lds_b128(
                (v4i*)(gwave + s * 4), (v4i*)(lwave + s * 4),
                /*offset=*/0, /*cpol=*/0);
        }
        asm volatile("s_wait_asynccnt 0" ::: "memory");

        // Each lane owns 4 rows = 36 contiguous floats (144 B, 16B aligned)
        // -> 9x ds_load_b128.
        f4 r[9];
        const f4* lp = (const f4*)(lwave + lane * 36);
#pragma unroll
        for (int i = 0; i < 9; ++i) r[i] = lp[i];

        const float* e = (const float*)&r[0];
        f4 o;
        o.x = det3x3(e + 0);
        o.y = det3x3(e + 9);
        o.z = det3x3(e + 18);
        o.w = det3x3(e + 27);

        // Streamed-once output: non-temporal 128-bit store.
        __builtin_nontemporal_store(o, (f4*)(out + blockRow0 + (long long)t * 4));
    } else {
        // ---------- tail path (not hit for N = 2^23, kept for safety) ----------
#pragma unroll
        for (int rr = 0; rr < ROWS_PER_THREAD; ++rr) {
            long long row = blockRow0 + (long long)t * ROWS_PER_THREAD + rr;
            if (row < nrows) {
                float m[9];
                const float* g = x + row * 9;
#pragma unroll
                for (int k = 0; k < 9; ++k) m[k] = g[k];
                out[row] = det3x3(m);
            }
        }
    }
}

extern "C" void kernel_launch(void* const* d_in, const int* in_sizes, int n_in,
                              void* d_out, int out_size, void* d_ws, size_t ws_size,
                              hipStream_t stream) {
    const float* x = (const float*)d_in[0];
    float* out = (float*)d_out;

    const long long nfloats = (long long)in_sizes[0];  // rows * 9
    const long long nrows = nfloats / 9;               // 8,388,608
    const int nblocks = (int)((nrows + ROWS_PER_BLOCK - 1) / ROWS_PER_BLOCK);

    det3_async_kernel<<<nblocks, BLOCK, 0, stream>>>(x, out, nrows);
}